// OriginTransformer_38620345925998
// MI455X (gfx1250) — compile-verified
//
#include <hip/hip_runtime.h>
#include <math.h>

#define B_  16
#define H_  2048
#define N_  1024
#define D_  2048

typedef __attribute__((ext_vector_type(2))) float v2f;
typedef __attribute__((ext_vector_type(4))) float v4f;
typedef __attribute__((ext_vector_type(8))) float v8f;

// ---------------------------------------------------------------------------
// Stage 1 (one block per batch b):
//   rowsum[d] = W[i1,d]+W[i2,d];  scores[h] = rowsum[z[h]] + rowsum[z[h-1]+N]
//   softmax over H (deterministic fixed-order reductions)
//   weighted_sum = deterministic scatter-add of attn (exclusive lane ownership)
//   combined     = weighted_sum + onehot(i1) + onehot(i2)
// ---------------------------------------------------------------------------
__global__ __launch_bounds__(128) void ot_stage1(const int* __restrict__ z,
                                                 const float* __restrict__ W,
                                                 float* __restrict__ wsum_g,   // [B][D]
                                                 float* __restrict__ comb_g)   // [B][D]
{
  __shared__ float rowsum[D_];
  __shared__ float sc[H_];
  __shared__ int   zb[H_];
  __shared__ float part[4][D_];
  __shared__ float redA[4];
  __shared__ float redB[4];

  const int b    = blockIdx.x;
  const int tid  = threadIdx.x;
  const int lane = tid & 31;
  const int w    = tid >> 5;
  const int* zrow = z + b * H_;

  const int i1 = zrow[H_ - 1];
  const int i2 = zrow[H_ - 2] + N_;

  for (int d = tid; d < D_; d += 128)
    rowsum[d] = W[(size_t)i1 * D_ + d] + W[(size_t)i2 * D_ + d];
  for (int h = tid; h < H_; h += 128)
    zb[h] = zrow[h];
  for (int idx = tid; idx < 4 * D_; idx += 128)
    (&part[0][0])[idx] = 0.0f;
  __syncthreads();

  // scores
  for (int h = tid; h < H_; h += 128) {
    float s = rowsum[zb[h]];
    if (h > 0) s += rowsum[zb[h - 1] + N_];
    sc[h] = s;
  }
  __syncthreads();

  // block max (deterministic order)
  float m = -1e30f;
  for (int h = tid; h < H_; h += 128) m = fmaxf(m, sc[h]);
  for (int off = 16; off; off >>= 1) m = fmaxf(m, __shfl_down(m, off, 32));
  if (lane == 0) redA[w] = m;
  __syncthreads();
  m = fmaxf(fmaxf(redA[0], redA[1]), fmaxf(redA[2], redA[3]));

  // block sum of exp
  float s = 0.0f;
  for (int h = tid; h < H_; h += 128) s += expf(sc[h] - m);
  for (int off = 16; off; off >>= 1) s += __shfl_down(s, off, 32);
  if (lane == 0) redB[w] = s;
  __syncthreads();
  s = redB[0] + redB[1] + redB[2] + redB[3];
  const float inv = 1.0f / s;

  for (int h = tid; h < H_; h += 128) sc[h] = expf(sc[h] - m) * inv;
  __syncthreads();

  // deterministic scatter: wave w scans h in [w*512, w*512+512);
  // lane L exclusively owns all d with (d & 31) == L  -> no atomics needed.
  {
    const int h0 = w * 512;
    int dprev = (h0 > 0) ? zb[h0 - 1] : 0;
    for (int i = 0; i < 512; ++i) {
      const int h  = h0 + i;
      const int d0 = zb[h];
      const float a = sc[h];
      if ((d0 & 31) == lane) part[w][d0] += a;
      if (h > 0) {
        const int dp = dprev + N_;
        if ((dp & 31) == lane) part[w][dp] += a;
      }
      dprev = d0;
    }
  }
  __syncthreads();

  for (int d = tid; d < D_; d += 128) {
    const float v = part[0][d] + part[1][d] + part[2][d] + part[3][d];
    wsum_g[(size_t)b * D_ + d] = v;
    comb_g[(size_t)b * D_ + d] = v + ((d == i1) ? 1.0f : 0.0f)
                                   + ((d == i2) ? 1.0f : 0.0f);
  }
}

// ---------------------------------------------------------------------------
// Stage 2: two (16 x 2048) x (2048 x 1024) GEMMs via V_WMMA_F32_16X16X4_F32.
// One BLOCK (8 waves) = one 16(batch) x 16(col) output tile; wave w contracts
// K in [256w, 256w+256), partials reduced through LDS in fixed order.
// b128 loads: lane-half 0 holds K [k..k+3], half 1 holds K [k+4..k+7];
// wmma #1 consumes elems (0,1) -> K {k,k+1,k+4,k+5},
// wmma #2 consumes elems (2,3) -> K {k+2,k+3,k+6,k+7}  (K-sum is commutative,
// and A/B use identical lane->K maps, so the contraction is exact).
//   A 16x4 layout: lanes 0-15 {K0,K1}, lanes 16-31 {K2,K3}  (M = lane&15)
//   B 4x16 layout: lanes 0-15 {K0,K1}, lanes 16-31 {K2,K3}  (N = lane&15)
//   C/D layout:    VGPR r, lane-half hf -> M = r + 8*hf, N = lane&15
// ---------------------------------------------------------------------------
__global__ __launch_bounds__(256) void ot_wmma_gemm(const float* __restrict__ wsum_g,
                                                    const float* __restrict__ comb_g,
                                                    const float* __restrict__ Vm,
                                                    const float* __restrict__ Fm,
                                                    float* __restrict__ out)
{
  __shared__ float red[8 * 256];

  const int tid  = threadIdx.x;
  const int lane = tid & 31;
  const int w    = tid >> 5;                 // K-chunk 0..7
  const int mat  = blockIdx.x >> 6;          // 0 -> xi_A (V), 1 -> xi_F (F)
  const int tile = blockIdx.x & 63;          // output column tile
  const int half = lane >> 4;
  const int lm   = lane & 15;

  const float* A  = mat ? comb_g : wsum_g;
  const float* Bm = mat ? Fm : Vm;

  const float* aPtr = A  + (size_t)lm * D_ + w * 256 + 4 * half;
  const float* bPtr = Bm + (size_t)(tile * 16 + lm) * D_ + w * 256 + 4 * half;

  v8f acc = {0.f, 0.f, 0.f, 0.f, 0.f, 0.f, 0.f, 0.f};

#pragma unroll 8
  for (int k = 0; k < 256; k += 8) {
    v4f a  = *(const v4f*)(aPtr + k);
    v4f bb = *(const v4f*)(bPtr + k);
    v2f a0 = {a.x, a.y};
    v2f a1 = {a.z, a.w};
    v2f b0 = {bb.x, bb.y};
    v2f b1 = {bb.z, bb.w};
    acc = __builtin_amdgcn_wmma_f32_16x16x4_f32(false, a0, false, b0,
                                                (short)0, acc, false, false);
    acc = __builtin_amdgcn_wmma_f32_16x16x4_f32(false, a1, false, b1,
                                                (short)0, acc, false, false);
  }

  // stash this wave's partial tile: red[w][m*16+n]
#pragma unroll
  for (int r = 0; r < 8; ++r)
    red[w * 256 + (r + 8 * half) * 16 + lm] = acc[r];
  __syncthreads();

  // fixed-order cross-wave reduction; thread t owns output element t = m*16+n
  const int m = tid >> 4;                    // batch (M)
  const int n = tid & 15;                    // column within tile (N)
  float v = 0.0f;
#pragma unroll
  for (int ww = 0; ww < 8; ++ww) v += red[ww * 256 + tid];

  out[(size_t)(mat ? 2 : 1) * (B_ * N_) + (size_t)m * N_ + (tile * 16 + n)] = v;
}

// ---------------------------------------------------------------------------
// Stage 3: xi = xi_A + xi_F
// ---------------------------------------------------------------------------
__global__ void ot_add(float* __restrict__ out)
{
  const int i = blockIdx.x * blockDim.x + threadIdx.x;
  if (i < B_ * N_)
    out[i] = out[B_ * N_ + i] + out[2 * B_ * N_ + i];
}

extern "C" void kernel_launch(void* const* d_in, const int* in_sizes, int n_in,
                              void* d_out, int out_size, void* d_ws, size_t ws_size,
                              hipStream_t stream)
{
  const int*   z  = (const int*)d_in[0];     // (B,H) int32
  const float* F  = (const float*)d_in[1];   // (D,D)
  const float* V  = (const float*)d_in[2];   // (D,D)
  const float* W  = (const float*)d_in[3];   // (D,D)
  float* out = (float*)d_out;                // [xi | xi_A | xi_F], each B*N

  float* wsum_g = (float*)d_ws;              // B*D floats
  float* comb_g = wsum_g + (size_t)B_ * D_;  // B*D floats (total 256 KB of ws)

  ot_stage1<<<B_, 128, 0, stream>>>(z, W, wsum_g, comb_g);
  ot_wmma_gemm<<<128, 256, 0, stream>>>(wsum_g, comb_g, V, F, out);
  ot_add<<<(B_ * N_ + 255) / 256, 256, 0, stream>>>(out);
}